// MumbaiTrafficGNN_68453188763742
// MI455X (gfx1250) — compile-verified
//
#include <hip/hip_runtime.h>
#include <hip/hip_bf16.h>

typedef __attribute__((ext_vector_type(2))) float v2f;
typedef __attribute__((ext_vector_type(8))) float v8f;

#define HID 64
#define BN_EPS 1e-5f

// ---------------------------------------------------------------------------
// WMMA fp32 GEMM: Out[M,64] = act(X[M,K] @ W[K,64] + bias), one wave per
// 16x16 tile, K accumulated 4 at a time with V_WMMA_F32_16X16X4_F32.
// fp32 A 16x4 layout: lanes 0-15 hold K={k0,k0+1}, lanes 16-31 K={k0+2,k0+3};
// B 4x16 mirrored; C/D: VGPR j, lane l -> M = j + 8*(l>>4), N = l&15.
// ---------------------------------------------------------------------------
__global__ void gcn_gemm_wmma(const float* __restrict__ X,
                              const float* __restrict__ W,
                              const float* __restrict__ bias,
                              float* __restrict__ Out,
                              int M, int K, int do_relu) {
    const int lane = threadIdx.x & 31;
    const int wave = threadIdx.x >> 5;
    const int tilesN = HID / 16;                  // 4
    const int nTiles = (M / 16) * tilesN;
    const int tile = blockIdx.x * (blockDim.x >> 5) + wave;  // wave-uniform
    if (tile >= nTiles) return;                   // uniform branch: EXEC stays all-1

    const int tm = (tile / tilesN) * 16;
    const int tn = (tile % tilesN) * 16;
    const int half = lane >> 4;                   // which K pair this lane holds
    const int l15  = lane & 15;

    v8f acc = {};
    for (int k0 = 0; k0 < K; k0 += 4) {
        const int ka = k0 + 2 * half;
        v2f a, b;
        const float* xr = X + (size_t)(tm + l15) * K + ka;
        a.x = xr[0];
        a.y = xr[1];
        const float* wr = W + (size_t)ka * HID + (tn + l15);
        b.x = wr[0];
        b.y = wr[HID];
        acc = __builtin_amdgcn_wmma_f32_16x16x4_f32(
            /*neg_a=*/false, a, /*neg_b=*/false, b,
            /*c_mod=*/(short)0, acc, /*reuse_a=*/false, /*reuse_b=*/false);
    }

    const int n = tn + l15;
    const float bv = bias[n];
#pragma unroll
    for (int j = 0; j < 8; ++j) {
        const int m = tm + j + 8 * half;
        float v = acc[j] + bv;
        if (do_relu) v = fmaxf(v, 0.0f);
        Out[(size_t)m * HID + n] = v;
    }
}

// ---------------------------------------------------------------------------
// Degree / normalization
// ---------------------------------------------------------------------------
__global__ void deg_init(float* __restrict__ deg, int N) {
    int i = blockIdx.x * blockDim.x + threadIdx.x;
    if (i < N) deg[i] = 1.0f;   // self-loop contribution
}

__global__ void deg_count(const int* __restrict__ ei, float* __restrict__ deg, int E) {
    int e = blockIdx.x * blockDim.x + threadIdx.x;
    if (e < E) atomicAdd(&deg[ei[E + e]], 1.0f);  // dst row of edge_index
}

__global__ void deg_to_dinv(float* __restrict__ deg, int N) {
    int i = blockIdx.x * blockDim.x + threadIdx.x;
    if (i < N) deg[i] = rsqrtf(fmaxf(deg[i], 1.0f));
}

// ---------------------------------------------------------------------------
// agg init with self-loop term (src=dst=i, norm=dinv[i]^2); also zeroes BN stats
// ---------------------------------------------------------------------------
__global__ void init_agg(const float* __restrict__ t, const float* __restrict__ dinv,
                         float* __restrict__ agg, float* __restrict__ stats, int N) {
    int g = blockIdx.x * blockDim.x + threadIdx.x;
    if (g < 2 * HID) stats[g] = 0.0f;
    if (g < N * HID) {
        int node = g >> 6;
        float di = dinv[node];
        agg[g] = t[g] * di * di;
    }
}

// ---------------------------------------------------------------------------
// Edge scatter: lane = feature so each edge's 64 floats are coalesced.
// ---------------------------------------------------------------------------
__global__ void edge_scatter(const int* __restrict__ ei, const float* __restrict__ t,
                             const float* __restrict__ dinv, float* __restrict__ agg,
                             int E) {
    int g = blockIdx.x * blockDim.x + threadIdx.x;
    int e = g >> 6;
    if (e >= E) return;
    int f = g & 63;
    int s = ei[e];
    int d = ei[E + e];
    float w = dinv[s] * dinv[d];
    atomicAdd(&agg[(size_t)d * HID + f], t[(size_t)s * HID + f] * w);
}

// ---------------------------------------------------------------------------
// BatchNorm stats: per-feature sum / sumsq with block-strided partials
// ---------------------------------------------------------------------------
__global__ void bn_stats(const float* __restrict__ agg, float* __restrict__ stats, int N) {
    int g = blockIdx.x * blockDim.x + threadIdx.x;
    int f = g & 63;
    int row0 = g >> 6;
    int stride = (gridDim.x * blockDim.x) >> 6;
    float s = 0.0f, q = 0.0f;
    for (int r = row0; r < N; r += stride) {
        float v = agg[(size_t)r * HID + f];
        s += v;
        q += v * v;
    }
    atomicAdd(&stats[f], s);
    atomicAdd(&stats[HID + f], q);
}

// ---------------------------------------------------------------------------
// BN apply + ReLU + residual (h updated in place)
// ---------------------------------------------------------------------------
__global__ void bn_apply(const float* __restrict__ agg, const float* __restrict__ stats,
                         const float* __restrict__ gamma, const float* __restrict__ beta,
                         float* __restrict__ h, int N) {
    int g = blockIdx.x * blockDim.x + threadIdx.x;
    if (g >= N * HID) return;
    int f = g & 63;
    float invN = 1.0f / (float)N;
    float mu = stats[f] * invN;
    float var = stats[HID + f] * invN - mu * mu;
    float rs = rsqrtf(var + BN_EPS);
    float v = (agg[g] - mu) * rs * gamma[f] + beta[f];
    h[g] = fmaxf(v, 0.0f) + h[g];
}

// ---------------------------------------------------------------------------
extern "C" void kernel_launch(void* const* d_in, const int* in_sizes, int n_in,
                              void* d_out, int out_size, void* d_ws, size_t ws_size,
                              hipStream_t stream) {
    const float* x       = (const float*)d_in[0];   // [N,32]
    const int*   ei      = (const int*)d_in[1];     // [2,E]
    const float* W_embed = (const float*)d_in[2];   // [32,64]
    const float* b_embed = (const float*)d_in[3];   // [64]
    const float* W_convs = (const float*)d_in[4];   // [3,64,64]
    const float* b_convs = (const float*)d_in[5];   // [3,64]
    const float* gamma   = (const float*)d_in[6];   // [64]
    const float* beta    = (const float*)d_in[7];   // [64]
    float* h = (float*)d_out;                       // [N,64] (holds h across layers)

    const int IN_F = 32;
    const int N = in_sizes[0] / IN_F;   // 50000
    const int E = in_sizes[1] / 2;      // 800000

    // Workspace layout (floats): dinv[N] | t[N*64] | agg[N*64] | stats[128]
    float* ws    = (float*)d_ws;
    float* dinv  = ws;
    float* t     = ws + ((N + 63) & ~63);
    float* agg   = t + (size_t)N * HID;
    float* stats = agg + (size_t)N * HID;

    const int B = 256;

    // 1) symmetric GCN normalization: deg -> rsqrt
    deg_init<<<(N + B - 1) / B, B, 0, stream>>>(dinv, N);
    deg_count<<<(E + B - 1) / B, B, 0, stream>>>(ei, dinv, E);
    deg_to_dinv<<<(N + B - 1) / B, B, 0, stream>>>(dinv, N);

    // 2) embed: h = relu(x @ W_embed + b_embed)   [WMMA fp32]
    const int nTiles = (N / 16) * (HID / 16);     // 12500
    const int wavesPerBlock = B / 32;
    gcn_gemm_wmma<<<(nTiles + wavesPerBlock - 1) / wavesPerBlock, B, 0, stream>>>(
        x, W_embed, b_embed, h, N, IN_F, /*relu=*/1);

    // 3) three GCNConv + BN + ReLU + residual layers
    for (int l = 0; l < 3; ++l) {
        gcn_gemm_wmma<<<(nTiles + wavesPerBlock - 1) / wavesPerBlock, B, 0, stream>>>(
            h, W_convs + (size_t)l * HID * HID, b_convs + (size_t)l * HID, t, N, HID,
            /*relu=*/0);
        init_agg<<<((size_t)N * HID + B - 1) / B, B, 0, stream>>>(t, dinv, agg, stats, N);
        edge_scatter<<<((size_t)E * HID + B - 1) / B, B, 0, stream>>>(ei, t, dinv, agg, E);
        bn_stats<<<256, B, 0, stream>>>(agg, stats, N);
        bn_apply<<<((size_t)N * HID + B - 1) / B, B, 0, stream>>>(agg, stats, gamma, beta, h, N);
    }
}